// PSP_29360396436106
// MI455X (gfx1250) — compile-verified
//
#include <hip/hip_runtime.h>

// PSP exponential scan  syn_t = 0.5*syn_{t-1} + 0.5*x_t  over T=64, as
// syns = L * X with L[t,s] = 2^-(t-s+1) (lower-triangular, Toeplitz, exact
// powers of two in f32), computed with V_WMMA_F32_16X16X4_F32 on gfx1250.

typedef float v2f __attribute__((ext_vector_type(2)));
typedef float v8f __attribute__((ext_vector_type(8)));

#define TT 64
#define WAVES_PER_BLOCK 8

__global__ __launch_bounds__(256) void psp_scan_wmma_kernel(
    const float* __restrict__ x, float* __restrict__ out, int NC) {
  const int lane = threadIdx.x & 31;
  const int wave = threadIdx.x >> 5;
  const int jblock = blockIdx.x * WAVES_PER_BLOCK + wave;
  const int col = (jblock << 4) + (lane & 15);
  const int hi = lane >> 4;  // lane half: selects K/M offset per WMMA layouts

  // ---- A tiles: L is Toeplitz, only 4 distinct 16x16 diagonal blocks.
  // a[d4][q] is a 16x4 f32 tile: row M = lane&15, K = q*4 + hi*2 + v.
  // coef(t,s) = 2^-(d+1) for d = t-s >= 0, else 0. Exact via bit pattern.
  v2f a[4][4];
  {
    const int r = lane & 15;
#pragma unroll
    for (int d4 = 0; d4 < 4; ++d4) {
#pragma unroll
      for (int q = 0; q < 4; ++q) {
#pragma unroll
        for (int v = 0; v < 2; ++v) {
          const int s4 = q * 4 + hi * 2 + v;   // k within the 16-block
          const int d = d4 * 16 + r - s4;      // t - s
          const int ebits = (127 - (d + 1)) << 23;  // 2^-(d+1), normal range
          a[d4][q][v] = (d >= 0) ? __int_as_float(ebits) : 0.0f;
        }
      }
    }
  }

  // ---- B tiles: x[s, col], s = 0..63. B 4x16 layout: N = lane&15,
  // K = v + 2*hi within each 4-row sub-tile.
  v2f b[4][4];
#pragma unroll
  for (int kb = 0; kb < 4; ++kb) {
#pragma unroll
    for (int q = 0; q < 4; ++q) {
#pragma unroll
      for (int v = 0; v < 2; ++v) {
        const int s = kb * 16 + q * 4 + hi * 2 + v;
        b[kb][q][v] = x[(size_t)s * (size_t)NC + (size_t)col];
      }
    }
  }

  // ---- Block-lower-triangular product: 10 blocks x 4 K-steps = 40 WMMAs.
  v8f acc[4] = {};
#pragma unroll
  for (int mb = 0; mb < 4; ++mb) {
#pragma unroll
    for (int kb = 0; kb < 4; ++kb) {
      if (kb <= mb) {
        const int d4 = mb - kb;
#pragma unroll
        for (int q = 0; q < 4; ++q) {
          acc[mb] = __builtin_amdgcn_wmma_f32_16x16x4_f32(
              /*neg_a=*/false, a[d4][q], /*neg_b=*/false, b[kb][q],
              /*c_mod=*/(short)0, acc[mb], /*reuse_a=*/false,
              /*reuse_b=*/false);
        }
      }
    }
  }

  // ---- Store D tiles: 16x16 f32 C/D layout: M = v + 8*hi, N = lane&15.
#pragma unroll
  for (int mb = 0; mb < 4; ++mb) {
#pragma unroll
    for (int v = 0; v < 8; ++v) {
      const int t = mb * 16 + v + hi * 8;
      out[(size_t)t * (size_t)NC + (size_t)col] = acc[mb][v];
    }
  }
}

extern "C" void kernel_launch(void* const* d_in, const int* in_sizes, int n_in,
                              void* d_out, int out_size, void* d_ws, size_t ws_size,
                              hipStream_t stream) {
  const float* x = (const float*)d_in[0];
  float* out = (float*)d_out;
  const int total = in_sizes[0];      // T*N*C = 64 * 512 * 2048
  const int NC = total / TT;          // 1,048,576 columns
  const int cols_per_block = 16 * WAVES_PER_BLOCK;  // 128
  const int grid = NC / cols_per_block;             // 8192 (exact)
  psp_scan_wmma_kernel<<<grid, 32 * WAVES_PER_BLOCK, 0, stream>>>(x, out, NC);
}